// TransformerEstimator_42468636623029
// MI455X (gfx1250) — compile-verified
//
#include <hip/hip_runtime.h>
#include <math.h>

// ---------------- problem constants ----------------
#define B_    16
#define TS_   1024
#define TP_   256
#define DIM_  256
#define CH_   512
#define H_    512
#define NPROWS (B_*TP_)        // 4096 prompt rows
#define NSROWS (B_*TS_)        // 16384 spec rows
#define NROWS  (NPROWS+NSROWS) // 20480 total rows

// ---------------- GEMM tile config ----------------
// Block tile 64x128, 8 waves in a 2x4 grid, each wave owns a 32x32 register
// tile (4 f32-WMMA accumulators) so A and B fragments are each reused twice.
// B tiles are streamed global->LDS by the Tensor Data Mover (double buffered).
#define KT 32
#define BM 64
#define BN 128
#define ASTRIDE (KT+1)    // 33, odd stride -> conflict-free A reads
#define BSTRIDE (BN+8)    // 136 dwords; TDM pads 8 dwords after each 128-dword row

typedef float v2f __attribute__((ext_vector_type(2)));
typedef float v8f __attribute__((ext_vector_type(8)));
typedef unsigned int u32x4 __attribute__((ext_vector_type(4)));
typedef unsigned int u32x8 __attribute__((ext_vector_type(8)));

// LDS byte offset of a __shared__ object (generic -> addrspace(3) -> int)
__device__ __forceinline__ unsigned lds_offset(const void* p) {
  return (unsigned)(unsigned long long)(__attribute__((address_space(3))) const char*)p;
}

// ---- Tensor Data Mover: async load of a KTxBN f32 tile into LDS ----
// Descriptor per CDNA5 ISA:
//  group0: [1:0]count=1 | [63:32]lds_addr | [120:64]global_addr | [127:126]type=2
//  group1: data_size=4B, pad 8 dwords after every 128 dwords (-> BSTRIDE=136),
//          tensor_dim0=BN, tensor_dim1=KT (tile always in-bounds), tile 128x32,
//          dim0 stride = ldb elements.
__device__ __forceinline__ void tdm_load_B(const float* gtile, unsigned ldb,
                                           unsigned lds_off) {
  unsigned long long ga = (unsigned long long)gtile;
  u32x4 g0;
  g0.x = 1u;                                   // count=1, user mode
  g0.y = lds_off;                              // LDS byte address
  g0.z = (unsigned)ga;                         // global addr [31:0]
  g0.w = (unsigned)(ga >> 32) | (2u << 30);    // global addr [56:32] | type=2
  u32x8 g1;
  g1.s0 = (2u << 16)        // data_size = 4 bytes
        | (1u << 20)        // pad_enable
        | (6u << 22)        // pad_interval: every 128 dwords
        | (7u << 25);       // pad_amount: 8 dwords
  g1.s1 = ((unsigned)BN) << 16;   // tensor_dim0 = 128 (bits 63:48)
  g1.s2 = ((unsigned)KT) << 16;   // tensor_dim1 = 32  (bits 95:80)
  g1.s3 = ((unsigned)BN) << 16;   // tile_dim0  = 128 (bits 127:112)
  g1.s4 = (unsigned)KT;           // tile_dim1  = 32  (bits 143:128)
  g1.s5 = ldb;                    // tensor_dim0_stride (elements)
  g1.s6 = 0u;
  g1.s7 = 0u;
  asm volatile("tensor_load_to_lds %0, %1" :: "s"(g0), "s"(g1) : "memory");
}

// ---------------- f32 WMMA 16x16x4 ----------------
// A 16x4 : VGPR0 = {lanes0-15: K=0, lanes16-31: K=2}, VGPR1 = {K=1, K=3}; M = lane&15
// B 4x16 : VGPR0 = {lanes0-15: K=0, lanes16-31: K=2}, VGPR1 = {K=1, K=3}; N = lane&15
// C/D    : VGPR i = {lanes0-15: M=i, lanes16-31: M=i+8}; N = lane&15
__device__ __forceinline__ v8f wmma4(v2f a, v2f b, v8f c) {
  return __builtin_amdgcn_wmma_f32_16x16x4_f32(false, a, false, b, (short)0, c,
                                               false, false);
}

// One KT-deep accumulation for a 32x32 per-wave tile:
//   c00=(m0,n0) c01=(m0,n0+64) c10=(m0+16,n0) c11=(m0+16,n0+64)
__device__ __forceinline__ void mma_chunk4(const float* Ash, const float* Bsh,
                                           int am0, int bn0, int lane,
                                           v8f& c00, v8f& c01, v8f& c10, v8f& c11) {
  const int half = lane >> 4;
  const int l = lane & 15;
  const float* arow0 = Ash + (size_t)(am0 + l) * ASTRIDE;
  const float* arow1 = arow0 + 16 * ASTRIDE;
#pragma unroll
  for (int k = 0; k < KT; k += 4) {
    const int kk = k + 2 * half;
    v2f a0, a1, b0, b1;
    a0.x = arow0[kk];
    a0.y = arow0[kk + 1];
    a1.x = arow1[kk];
    a1.y = arow1[kk + 1];
    b0.x = Bsh[(size_t)kk * BSTRIDE + bn0 + l];
    b0.y = Bsh[(size_t)(kk + 1) * BSTRIDE + bn0 + l];
    b1.x = Bsh[(size_t)kk * BSTRIDE + bn0 + 64 + l];
    b1.y = Bsh[(size_t)(kk + 1) * BSTRIDE + bn0 + 64 + l];
    c00 = wmma4(a0, b0, c00);
    c01 = wmma4(a0, b1, c01);
    c10 = wmma4(a1, b0, c10);
    c11 = wmma4(a1, b1, c11);
  }
}

// stage BMxKT A tile, source row-major [M][K] (already offset to m0,k0)
__device__ __forceinline__ void stage_A_rowmajor(float* Ash, const float* G,
                                                 size_t lda, int tid) {
#pragma unroll
  for (int r = 0; r < (BM * KT) / 256; ++r) {
    int idx = tid + r * 256;
    int m = idx >> 5, k = idx & (KT - 1);
    Ash[m * ASTRIDE + k] = G[(size_t)m * lda + k];
  }
}

// stage BMxKT A tile, source K-major [K][M] (already offset to k0,t0): transpose in LDS
__device__ __forceinline__ void stage_A_colmajor(float* Ash, const float* G,
                                                 size_t ldg, int tid) {
#pragma unroll
  for (int r = 0; r < (BM * KT) / 256; ++r) {
    int idx = tid + r * 256;
    int c = idx >> 6, t = idx & (BM - 1);
    Ash[t * ASTRIDE + c] = G[(size_t)c * ldg + t];
  }
}

// ---------------- K0: sequence lengths from masks ----------------
__global__ __launch_bounds__(256) void len_kernel(const int* __restrict__ xm,
                                                  const int* __restrict__ pm,
                                                  int* __restrict__ lens) {
  __shared__ int red[256];
  const int b = blockIdx.x, tid = threadIdx.x;
  int s = 0;
  for (int t = tid; t < TS_; t += 256) s += xm[(size_t)b * TS_ + t];
  red[tid] = s;
  __syncthreads();
  for (int o = 128; o > 0; o >>= 1) {
    if (tid < o) red[tid] += red[tid + o];
    __syncthreads();
  }
  if (tid == 0) lens[B_ + b] = TS_ - red[0];  // s_len
  __syncthreads();
  red[tid] = (tid < TP_) ? pm[(size_t)b * TP_ + tid] : 0;
  __syncthreads();
  for (int o = 128; o > 0; o >>= 1) {
    if (tid < o) red[tid] += red[tid + o];
    __syncthreads();
  }
  if (tid == 0) lens[b] = TP_ - red[0];  // p_len
}

// ---------------- K1: sinusoidal + MLP step embeddings ----------------
// emb layout: [0][b][H] = d_emb, [1][b][H] = p_emb
__global__ __launch_bounds__(256) void emb_kernel(
    const float* __restrict__ dstep, const float* __restrict__ pstep,
    const float* __restrict__ w1, const float* __restrict__ b1,
    const float* __restrict__ w2, const float* __restrict__ b2,
    float* __restrict__ emb) {
  __shared__ float e[DIM_];
  __shared__ float h1[DIM_ * 4];
  const int b = blockIdx.x & (B_ - 1);
  const int which = blockIdx.x >> 4;  // 0 = diffusion, 1 = prompt
  const int tid = threadIdx.x;
  const float t = which ? pstep[b] : dstep[b];
  {
    const int half = DIM_ / 2;  // 128
    if (tid < half) {
      float fr = expf(-9.2103403719761836f * (float)tid / (float)(half - 1));
      float a = t * fr;
      e[tid] = sinf(a);
      e[tid + half] = cosf(a);
    }
  }
  __syncthreads();
  for (int j = tid; j < DIM_ * 4; j += 256) {
    float acc = b1[j];
    for (int c = 0; c < DIM_; ++c) acc = fmaf(e[c], w1[(size_t)c * (DIM_ * 4) + j], acc);
    float sp = (acc > 20.f) ? acc : log1pf(expf(acc));  // softplus
    h1[j] = acc * tanhf(sp);                            // mish
  }
  __syncthreads();
  for (int k = tid; k < H_; k += 256) {
    float acc = b2[k];
    for (int c = 0; c < DIM_ * 4; ++c) acc = fmaf(h1[c], w2[(size_t)c * H_ + k], acc);
    emb[((size_t)which * B_ + b) * H_ + k] = acc;
  }
}

// ---------------- K2: prompt rows -> vAll[0 .. 4096) ----------------
__global__ __launch_bounds__(256) void gemm_prompt_kernel(
    const float* __restrict__ prompt, const float* __restrict__ pw,
    const float* __restrict__ pb, const float* __restrict__ emb,
    float* __restrict__ vAll) {
  __shared__ float Ash[BM * ASTRIDE];
  __shared__ float Bsh[2][KT * BSTRIDE];
  const int b = blockIdx.z, m0 = blockIdx.y * BM, n0 = blockIdx.x * BN;
  const int tid = threadIdx.x, lane = tid & 31, wave = tid >> 5;
  const int wm0 = (wave >> 2) * 32, wn0 = (wave & 3) * 16;
  const unsigned boff[2] = {lds_offset(&Bsh[0][0]), lds_offset(&Bsh[1][0])};
  const float* Ag = prompt + ((size_t)b * TP_ + m0) * DIM_;
  const int nch = DIM_ / KT;  // 8
  v8f c00 = {}, c01 = {}, c10 = {}, c11 = {};
  if (wave == 0) tdm_load_B(pw + n0, H_, boff[0]);
  for (int i = 0; i < nch; ++i) {
    __syncthreads();  // Ash + Bsh[(i+1)&1] consumers done
    if (wave == 0) {
      if (i + 1 < nch) {
        tdm_load_B(pw + (size_t)(i + 1) * KT * H_ + n0, H_, boff[(i + 1) & 1]);
        __builtin_amdgcn_s_wait_tensorcnt((short)1);  // chunk i resident
      } else {
        __builtin_amdgcn_s_wait_tensorcnt((short)0);
      }
    }
    stage_A_rowmajor(Ash, Ag + (size_t)i * KT, DIM_, tid);
    __syncthreads();
    mma_chunk4(Ash, Bsh[i & 1], wm0, wn0, lane, c00, c01, c10, c11);
  }
  const int half = lane >> 4, l = lane & 15;
  const int na = n0 + wn0 + l, nb = na + 64;
  const float adda = pb[na] + emb[((size_t)B_ + b) * H_ + na];  // p_emb
  const float addb = pb[nb] + emb[((size_t)B_ + b) * H_ + nb];
  float* C = vAll + ((size_t)b * TP_ + m0) * H_;
#pragma unroll
  for (int i = 0; i < 8; ++i) {
    int mA = wm0 + i + 8 * half;
    int mB = mA + 16;
    C[(size_t)mA * H_ + na] = c00[i] + adda;
    C[(size_t)mA * H_ + nb] = c01[i] + addb;
    C[(size_t)mB * H_ + na] = c10[i] + adda;
    C[(size_t)mB * H_ + nb] = c11[i] + addb;
  }
}

// ---------------- K3: spec rows -> vAll[4096 .. 20480) ----------------
__global__ __launch_bounds__(256) void gemm_spec_kernel(
    const float* __restrict__ spec, const float* __restrict__ cond,
    const float* __restrict__ sw, const float* __restrict__ sb,
    const float* __restrict__ cw, const float* __restrict__ cb,
    const float* __restrict__ emb, float* __restrict__ vAll) {
  __shared__ float Ash[BM * ASTRIDE];
  __shared__ float Bsh[2][KT * BSTRIDE];
  const int b = blockIdx.z, t0 = blockIdx.y * BM, n0 = blockIdx.x * BN;
  const int tid = threadIdx.x, lane = tid & 31, wave = tid >> 5;
  const int wm0 = (wave >> 2) * 32, wn0 = (wave & 3) * 16;
  const unsigned boff[2] = {lds_offset(&Bsh[0][0]), lds_offset(&Bsh[1][0])};
  const float* Ag1 = spec + (size_t)b * DIM_ * TS_ + t0;
  const float* Ag2 = cond + (size_t)b * CH_ * TS_ + t0;
  const int nspec = DIM_ / KT;          // 8
  const int nch = nspec + CH_ / KT;     // 24
#define SPEC_BSRC(i) ((i) < nspec ? sw + (size_t)(i) * KT * H_ + n0 \
                                  : cw + (size_t)((i) - nspec) * KT * H_ + n0)
#define SPEC_ASRC(i) ((i) < nspec ? Ag1 + (size_t)(i) * KT * TS_ \
                                  : Ag2 + (size_t)((i) - nspec) * KT * TS_)
  v8f c00 = {}, c01 = {}, c10 = {}, c11 = {};
  if (wave == 0) tdm_load_B(SPEC_BSRC(0), H_, boff[0]);
  for (int i = 0; i < nch; ++i) {
    __syncthreads();
    if (wave == 0) {
      if (i + 1 < nch) {
        tdm_load_B(SPEC_BSRC(i + 1), H_, boff[(i + 1) & 1]);
        __builtin_amdgcn_s_wait_tensorcnt((short)1);
      } else {
        __builtin_amdgcn_s_wait_tensorcnt((short)0);
      }
    }
    stage_A_colmajor(Ash, SPEC_ASRC(i), TS_, tid);
    __syncthreads();
    mma_chunk4(Ash, Bsh[i & 1], wm0, wn0, lane, c00, c01, c10, c11);
  }
#undef SPEC_BSRC
#undef SPEC_ASRC
  const int half = lane >> 4, l = lane & 15;
  const int na = n0 + wn0 + l, nb = na + 64;
  const float adda = sb[na] + cb[na] + emb[(size_t)b * H_ + na];  // d_emb
  const float addb = sb[nb] + cb[nb] + emb[(size_t)b * H_ + nb];
  float* C = vAll + ((size_t)NPROWS + (size_t)b * TS_ + t0) * H_;
#pragma unroll
  for (int i = 0; i < 8; ++i) {
    int mA = wm0 + i + 8 * half;
    int mB = mA + 16;
    C[(size_t)mA * H_ + na] = c00[i] + adda;
    C[(size_t)mA * H_ + nb] = c01[i] + addb;
    C[(size_t)mB * H_ + na] = c10[i] + adda;
    C[(size_t)mB * H_ + nb] = c11[i] + addb;
  }
}

// ---------------- K4: y = tanh(v @ dec_w + dec_b) ----------------
__global__ __launch_bounds__(256) void dec_kernel(
    const float* __restrict__ v, const float* __restrict__ dw,
    const float* __restrict__ db, float* __restrict__ y) {
  __shared__ float Ash[BM * ASTRIDE];
  __shared__ float Bsh[2][KT * BSTRIDE];
  const int m0 = blockIdx.y * BM, n0 = blockIdx.x * BN;
  const int tid = threadIdx.x, lane = tid & 31, wave = tid >> 5;
  const int wm0 = (wave >> 2) * 32, wn0 = (wave & 3) * 16;
  const unsigned boff[2] = {lds_offset(&Bsh[0][0]), lds_offset(&Bsh[1][0])};
  const int nch = H_ / KT;  // 16
  v8f c00 = {}, c01 = {}, c10 = {}, c11 = {};
  if (wave == 0) tdm_load_B(dw + n0, H_, boff[0]);
  for (int i = 0; i < nch; ++i) {
    __syncthreads();
    if (wave == 0) {
      if (i + 1 < nch) {
        tdm_load_B(dw + (size_t)(i + 1) * KT * H_ + n0, H_, boff[(i + 1) & 1]);
        __builtin_amdgcn_s_wait_tensorcnt((short)1);
      } else {
        __builtin_amdgcn_s_wait_tensorcnt((short)0);
      }
    }
    stage_A_rowmajor(Ash, v + (size_t)m0 * H_ + (size_t)i * KT, H_, tid);
    __syncthreads();
    mma_chunk4(Ash, Bsh[i & 1], wm0, wn0, lane, c00, c01, c10, c11);
  }
  const int half = lane >> 4, l = lane & 15;
  const int na = n0 + wn0 + l, nb = na + 64;
  const float dba = db[na], dbb = db[nb];
  float* C = y + (size_t)m0 * H_;
#pragma unroll
  for (int i = 0; i < 8; ++i) {
    int mA = wm0 + i + 8 * half;
    int mB = mA + 16;
    C[(size_t)mA * H_ + na] = tanhf(c00[i] + dba);
    C[(size_t)mA * H_ + nb] = tanhf(c01[i] + dbb);
    C[(size_t)mB * H_ + na] = tanhf(c10[i] + dba);
    C[(size_t)mB * H_ + nb] = tanhf(c11[i] + dbb);
  }
}

// ---------------- K5: o = y @ out_w + out_b, masked scatter ----------------
// d_out = [prompt_ret (B,TP,DIM)] ++ [spec_out (B,DIM,TS)]
__device__ __forceinline__ void scatter_out(float* __restrict__ out,
                                            const int* __restrict__ lens,
                                            int m, int n, float val) {
  if (m < NPROWS) {
    int b = m >> 8, tp = m & (TP_ - 1);
    bool ok = tp < lens[b];
    out[((size_t)b * TP_ + tp) * DIM_ + n] = ok ? val : 0.f;
  } else {
    int rs = m - NPROWS;
    int b = rs >> 10, t = rs & (TS_ - 1);
    bool ok = t < lens[B_ + b];
    // transpose into (B, DIM, TS)
    out[(size_t)NPROWS * DIM_ + ((size_t)b * DIM_ + n) * TS_ + t] = ok ? val : 0.f;
  }
}

__global__ __launch_bounds__(256) void out_kernel(
    const float* __restrict__ y, const float* __restrict__ ow,
    const float* __restrict__ ob, const int* __restrict__ lens,
    float* __restrict__ out) {
  __shared__ float Ash[BM * ASTRIDE];
  __shared__ float Bsh[2][KT * BSTRIDE];
  const int m0 = blockIdx.y * BM, n0 = blockIdx.x * BN;
  const int tid = threadIdx.x, lane = tid & 31, wave = tid >> 5;
  const int wm0 = (wave >> 2) * 32, wn0 = (wave & 3) * 16;
  const unsigned boff[2] = {lds_offset(&Bsh[0][0]), lds_offset(&Bsh[1][0])};
  const int nch = H_ / KT;  // 16
  v8f c00 = {}, c01 = {}, c10 = {}, c11 = {};
  if (wave == 0) tdm_load_B(ow + n0, DIM_, boff[0]);
  for (int i = 0; i < nch; ++i) {
    __syncthreads();
    if (wave == 0) {
      if (i + 1 < nch) {
        tdm_load_B(ow + (size_t)(i + 1) * KT * DIM_ + n0, DIM_, boff[(i + 1) & 1]);
        __builtin_amdgcn_s_wait_tensorcnt((short)1);
      } else {
        __builtin_amdgcn_s_wait_tensorcnt((short)0);
      }
    }
    stage_A_rowmajor(Ash, y + (size_t)m0 * H_ + (size_t)i * KT, H_, tid);
    __syncthreads();
    mma_chunk4(Ash, Bsh[i & 1], wm0, wn0, lane, c00, c01, c10, c11);
  }
  const int half = lane >> 4, l = lane & 15;
  const int na = n0 + wn0 + l, nb = na + 64;
  const float oba = ob[na], obb = ob[nb];
#pragma unroll
  for (int i = 0; i < 8; ++i) {
    int mA = m0 + wm0 + i + 8 * half;  // 64-row tiles never straddle P/S (4096 % 64 == 0)
    int mB = mA + 16;
    scatter_out(out, lens, mA, na, c00[i] + oba);
    scatter_out(out, lens, mA, nb, c01[i] + obb);
    scatter_out(out, lens, mB, na, c10[i] + oba);
    scatter_out(out, lens, mB, nb, c11[i] + obb);
  }
}

// ---------------- launcher ----------------
extern "C" void kernel_launch(void* const* d_in, const int* in_sizes, int n_in,
                              void* d_out, int out_size, void* d_ws, size_t ws_size,
                              hipStream_t stream) {
  (void)in_sizes; (void)n_in; (void)out_size; (void)ws_size;
  const float* spec        = (const float*)d_in[0];
  const int*   x_mask      = (const int*)d_in[1];
  const float* dstep       = (const float*)d_in[2];
  const float* pstep       = (const float*)d_in[3];
  const float* cond        = (const float*)d_in[4];
  /* d_in[5] = spk, unused by reference */
  const float* prompt      = (const float*)d_in[6];
  const int*   prompt_mask = (const int*)d_in[7];
  const float* w1 = (const float*)d_in[8];
  const float* b1 = (const float*)d_in[9];
  const float* w2 = (const float*)d_in[10];
  const float* b2 = (const float*)d_in[11];
  const float* sw = (const float*)d_in[12];
  const float* sb = (const float*)d_in[13];
  const float* cw = (const float*)d_in[14];
  const float* cb = (const float*)d_in[15];
  const float* pw = (const float*)d_in[16];
  const float* pb = (const float*)d_in[17];
  const float* dw = (const float*)d_in[18];
  const float* db = (const float*)d_in[19];
  const float* ow = (const float*)d_in[20];
  const float* ob = (const float*)d_in[21];

  // workspace layout (256B aligned): lens | emb | vAll | yAll  (~84 MB)
  char* ws = (char*)d_ws;
  int*   lens = (int*)ws;                                   // 2*B ints
  float* emb  = (float*)(ws + 256);                         // 2*B*H floats
  float* vAll = (float*)(ws + 256 + (size_t)2 * B_ * H_ * sizeof(float));
  float* yAll = vAll + (size_t)NROWS * H_;

  len_kernel<<<B_, 256, 0, stream>>>(x_mask, prompt_mask, lens);
  emb_kernel<<<2 * B_, 256, 0, stream>>>(dstep, pstep, w1, b1, w2, b2, emb);
  gemm_prompt_kernel<<<dim3(H_ / BN, TP_ / BM, B_), 256, 0, stream>>>(prompt, pw, pb, emb, vAll);
  gemm_spec_kernel<<<dim3(H_ / BN, TS_ / BM, B_), 256, 0, stream>>>(spec, cond, sw, sb, cw, cb, emb, vAll);
  dec_kernel<<<dim3(H_ / BN, NROWS / BM), 256, 0, stream>>>(vAll, dw, db, yAll);
  out_kernel<<<dim3(DIM_ / BN, NROWS / BM), 256, 0, stream>>>(yAll, ow, ob, lens, (float*)d_out);
}